// Quantizer_35536559407233
// MI455X (gfx1250) — compile-verified
//
#include <hip/hip_runtime.h>
#include <hip/hip_bf16.h>
#include <stdint.h>

#define NUM_CAND     80
#define CAND_CHUNK   4
#define QMAXF        255.0f
#define TILE         4096          // fp32 elements per TDM tile (16 KB)
#define MSE_BLOCKS   1024
#define MSE_THREADS  256
#define STR_BLOCKS   2048
#define STR_THREADS  256

typedef float v2f  __attribute__((ext_vector_type(2)));
typedef float v8f  __attribute__((ext_vector_type(8)));
typedef unsigned int u32x4 __attribute__((ext_vector_type(4)));
typedef int   i32x4 __attribute__((ext_vector_type(4)));
typedef int   i32x8 __attribute__((ext_vector_type(8)));

#if __has_builtin(__builtin_amdgcn_tensor_load_to_lds) && __has_builtin(__builtin_amdgcn_s_wait_tensorcnt)
#define HAVE_TDM 1
#else
#define HAVE_TDM 0
#endif
#if __has_builtin(__builtin_amdgcn_wmma_f32_16x16x4_f32)
#define HAVE_WMMA 1
#else
#define HAVE_WMMA 0
#endif

#if __has_builtin(__builtin_amdgcn_sched_barrier)
#define SCHED_FENCE() __builtin_amdgcn_sched_barrier(0)
#else
#define SCHED_FENCE()
#endif

// ---------- monotone int key for deterministic float min/max atomics ----------
__device__ __forceinline__ int f32_key(float x) {
    int i = __float_as_int(x);
    return i >= 0 ? i : (i ^ 0x7fffffff);
}
__device__ __forceinline__ float key_f32(int k) {
    int i = k >= 0 ? k : (k ^ 0x7fffffff);
    return __int_as_float(i);
}

// ---------- kernels ----------
__global__ void k_init(int* keys, float* mses) {
    if (threadIdx.x == 0) { keys[0] = 0x7fffffff; keys[1] = (int)0x80000000; }
    if (threadIdx.x < NUM_CAND) mses[threadIdx.x] = 0.0f;
}

__global__ __launch_bounds__(STR_THREADS)
void k_minmax(const float* __restrict__ x, long long n, int* keys) {
    float mn = __int_as_float(0x7f800000);   // +inf
    float mx = -mn;
    long long gid = (long long)blockIdx.x * blockDim.x + threadIdx.x;
    long long gs  = (long long)gridDim.x * blockDim.x;
    long long n4  = n >> 2;
    const float4* x4 = (const float4*)x;
    for (long long i = gid; i < n4; i += gs) {
        float4 v = x4[i];
        mn = fminf(mn, fminf(fminf(v.x, v.y), fminf(v.z, v.w)));
        mx = fmaxf(mx, fmaxf(fmaxf(v.x, v.y), fmaxf(v.z, v.w)));
    }
    for (long long i = (n4 << 2) + gid; i < n; i += gs) {
        float v = x[i]; mn = fminf(mn, v); mx = fmaxf(mx, v);
    }
#pragma unroll
    for (int off = 16; off; off >>= 1) {
        mn = fminf(mn, __shfl_xor(mn, off, 32));
        mx = fmaxf(mx, __shfl_xor(mx, off, 32));
    }
    __shared__ float smn[STR_THREADS / 32], smx[STR_THREADS / 32];
    int wid = threadIdx.x >> 5;
    if ((threadIdx.x & 31) == 0) { smn[wid] = mn; smx[wid] = mx; }
    __syncthreads();
    if (threadIdx.x == 0) {
        for (int w = 1; w < (int)(blockDim.x >> 5); ++w) {
            mn = fminf(mn, smn[w]); mx = fmaxf(mx, smx[w]);
        }
        atomicMin(&keys[0], f32_key(mn));
        atomicMax(&keys[1], f32_key(mx));
    }
}

// params[c] = {inv_scale, scale, lo=-offset, hi=255-offset}
__global__ void k_params(const int* __restrict__ keys, float4* __restrict__ params) {
    int c = threadIdx.x;
    if (c >= NUM_CAND) return;
    float xmin  = key_f32(keys[0]);
    float xmax  = key_f32(keys[1]);
    float range = xmax - xmin;
    float scale = (range - 0.01f * (float)c) / QMAXF;
    float inv   = 1.0f / scale;
    float off   = rintf(-xmin * inv);        // jnp.round == rint (half-to-even)
    params[c] = make_float4(inv, scale, -off, QMAXF - off);
}

__device__ __forceinline__ float qacc1(float x, float4 p, float a) {
    float t = rintf(x * p.x);                         // v_rndne_f32
    t = __builtin_amdgcn_fmed3f(t, p.z, p.w);         // clamp via v_med3_f32 (lo<=hi)
    float e = fmaf(t, p.y, -x);
    return fmaf(e, e, a);
}
__device__ __forceinline__ float qacc4(float4 v, float4 p, float a) {
    a = qacc1(v.x, p, a); a = qacc1(v.y, p, a);
    a = qacc1(v.z, p, a); a = qacc1(v.w, p, a);
    return a;
}

#if HAVE_TDM
// Issue a 1-D TDM tile load: TILE fp32 from gptr -> LDS byte offset lds_byte.
// D# per CDNA5 ISA ch.8: group0 {count=1, lds_addr, global_addr, type=2},
// group1 {data_size=4B, tensor_dim0=tile_dim0=TILE, tensor_dim1=1}.
__device__ __forceinline__ void tdm_tile(unsigned int lds_byte, const float* gptr) {
    uint64_t ga = (uint64_t)(uintptr_t)gptr;
    u32x4 g0 = { 1u,                                   // count=1, is_restore=0, gather off
                 lds_byte,                             // lds_addr [63:32]
                 (unsigned int)ga,                     // global_addr [95:64]
                 (((unsigned int)(ga >> 32)) & 0x01ffffffu) | (2u << 30) }; // ga[56:32] | type=2
    i32x8 g1 = { (int)0x00020000,                      // data_size=2 (4B), mask/pad/iterate=0
                 (int)((TILE & 0xffff) << 16),         // tensor_dim0[15:0]
                 (int)(((TILE >> 16) & 0xffff) | (1 << 16)), // tensor_dim0[31:16], tensor_dim1=1
                 (int)((TILE & 0xffff) << 16),         // tile_dim0
                 0,                                    // tile_dim1=0 (1-D), tile_dim2=0
                 (int)TILE,                            // tensor_dim0_stride[31:0]
                 0, 0 };
    i32x4 z4 = { 0, 0, 0, 0 };
    i32x8 z8 = { 0, 0, 0, 0, 0, 0, 0, 0 };
    __builtin_amdgcn_tensor_load_to_lds(g0, g1, z4, z4, z8, 0);
}
#endif

// Accumulate 8 elements (2 float4) into acc[NUM_CAND].
// Fences every CAND_CHUNK candidates stop the scheduler from hoisting all 80
// ds_load_b128 param fetches at once (that hoisting is what pushed allocation
// past 256 VGPRs -> vgpr-msb + occupancy loss). In-flight now: 4 params
// (16 VGPRs) + 80 acc + 8 elems + temps ~= 130-160 VGPRs.
__device__ __forceinline__ void half_accum(float4 va, float4 vb,
                                           const float4* __restrict__ lds_p,
                                           float* __restrict__ acc) {
#pragma unroll
    for (int cc = 0; cc < NUM_CAND; cc += CAND_CHUNK) {
        SCHED_FENCE();
#pragma unroll
        for (int c = cc; c < cc + CAND_CHUNK; ++c) {
            float4 p = lds_p[c];
            float a = acc[c];
            a = qacc4(va, p, a);
            a = qacc4(vb, p, a);
            acc[c] = a;
        }
    }
    SCHED_FENCE();
}

__device__ __forceinline__ void tile_accum(const float* __restrict__ lx, int tid,
                                           const float4* __restrict__ lds_p,
                                           float* __restrict__ acc) {
    const float4* lb = (const float4*)lx;
    half_accum(lb[tid], lb[tid + MSE_THREADS], lds_p, acc);
    half_accum(lb[tid + 2 * MSE_THREADS], lb[tid + 3 * MSE_THREADS], lds_p, acc);
}

__global__ __launch_bounds__(MSE_THREADS)
void k_mse(const float* __restrict__ x, long long n,
           const float4* __restrict__ params, float* __restrict__ mses) {
    __shared__ float  lds_x[2][TILE];
    __shared__ float4 lds_p[NUM_CAND];
    __shared__ float  lds_m[NUM_CAND];

    const int tid = threadIdx.x;
    if (tid < NUM_CAND) { lds_p[tid] = params[tid]; lds_m[tid] = 0.0f; }
    __syncthreads();

    float acc[NUM_CAND];
#pragma unroll
    for (int c = 0; c < NUM_CAND; ++c) acc[c] = 0.0f;

    const int ntiles = (int)(n / TILE);

#if HAVE_TDM
    const bool w0 = (tid < 32);                 // wave 0 drives the TDM
    int buf = 0;
    int t = blockIdx.x;
    if (w0 && t < ntiles)
        tdm_tile((unsigned int)(uintptr_t)&lds_x[0][0], x + (size_t)t * TILE);
    for (; t < ntiles; t += gridDim.x) {
        int nt = t + (int)gridDim.x;
        if (w0) {
            if (nt < ntiles) {                  // prefetch next tile, wait for current
                tdm_tile((unsigned int)(uintptr_t)&lds_x[buf ^ 1][0], x + (size_t)nt * TILE);
                __builtin_amdgcn_s_wait_tensorcnt(1);
            } else {
                __builtin_amdgcn_s_wait_tensorcnt(0);
            }
        }
        __syncthreads();
        tile_accum(&lds_x[buf][0], tid, lds_p, acc);
        __syncthreads();
        buf ^= 1;
    }
#else
    for (int t = blockIdx.x; t < ntiles; t += gridDim.x) {
        const float4* g4 = (const float4*)(x + (size_t)t * TILE);
        float4* l4 = (float4*)&lds_x[0][0];
        for (int i = tid; i < TILE / 4; i += MSE_THREADS) l4[i] = g4[i];
        __syncthreads();
        tile_accum(&lds_x[0][0], tid, lds_p, acc);
        __syncthreads();
    }
#endif

    // tail elements (n % TILE) via plain grid-stride loads.
    // Same chunked fences: without them this (rarely-executed) loop hoists all
    // 80 param loads and its liveness alone forces the whole kernel to ~500
    // VGPRs, wrecking hot-loop occupancy.
    long long tail = (long long)ntiles * TILE;
    long long gid  = (long long)blockIdx.x * blockDim.x + tid;
    long long gs   = (long long)gridDim.x * blockDim.x;
    for (long long i = tail + gid; i < n; i += gs) {
        float v = x[i];
#pragma unroll
        for (int cc = 0; cc < NUM_CAND; cc += CAND_CHUNK) {
            SCHED_FENCE();
#pragma unroll
            for (int c = cc; c < cc + CAND_CHUNK; ++c)
                acc[c] = qacc1(v, lds_p[c], acc[c]);
        }
        SCHED_FENCE();
    }

    // ---- cross-lane reduction of acc[c] via the matrix unit ----
    // A(16x4): vgpr0 = acc(lane), vgpr1 = 0  => row m sums acc[lane m]+acc[lane m+16]
    // B(4x16) = ones => every D column holds the row sums; 8 adds + 1 shfl finishes.
#if HAVE_WMMA
    v2f bones; bones[0] = 1.0f; bones[1] = 1.0f;
#pragma unroll
    for (int c = 0; c < NUM_CAND; ++c) {
        v2f a; a[0] = acc[c]; a[1] = 0.0f;
        v8f d = {};
        d = __builtin_amdgcn_wmma_f32_16x16x4_f32(false, a, false, bones,
                                                  (short)0, d, false, false);
        float s = ((d[0] + d[1]) + (d[2] + d[3])) + ((d[4] + d[5]) + (d[6] + d[7]));
        s += __shfl_xor(s, 16, 32);
        if ((tid & 31) == 0) atomicAdd(&lds_m[c], s);
    }
#else
#pragma unroll
    for (int c = 0; c < NUM_CAND; ++c) {
        float s = acc[c];
        for (int off = 16; off; off >>= 1) s += __shfl_xor(s, off, 32);
        if ((tid & 31) == 0) atomicAdd(&lds_m[c], s);
    }
#endif
    __syncthreads();
    if (tid < NUM_CAND) atomicAdd(&mses[tid], lds_m[tid]);
}

__global__ void k_best(const float* __restrict__ mses,
                       const float4* __restrict__ params, float4* __restrict__ best) {
    if (threadIdx.x == 0 && blockIdx.x == 0) {
        int b = 0; float bm = mses[0];
        for (int i = 1; i < NUM_CAND; ++i) {
            float m = mses[i];
            if (m < bm) { bm = m; b = i; }   // strict <: first minimum (matches argmin)
        }
        best[0] = params[b];
    }
}

__global__ __launch_bounds__(STR_THREADS)
void k_apply(const float* __restrict__ x, float* __restrict__ out,
             long long n, const float4* __restrict__ best) {
    float4 p = best[0];
    long long gid = (long long)blockIdx.x * blockDim.x + threadIdx.x;
    long long gs  = (long long)gridDim.x * blockDim.x;
    long long n4  = n >> 2;
    const float4* x4 = (const float4*)x;
    float4* o4 = (float4*)out;
    for (long long i = gid; i < n4; i += gs) {
        float4 v = x4[i], r;
        r.x = __builtin_amdgcn_fmed3f(rintf(v.x * p.x), p.z, p.w) * p.y;
        r.y = __builtin_amdgcn_fmed3f(rintf(v.y * p.x), p.z, p.w) * p.y;
        r.z = __builtin_amdgcn_fmed3f(rintf(v.z * p.x), p.z, p.w) * p.y;
        r.w = __builtin_amdgcn_fmed3f(rintf(v.w * p.x), p.z, p.w) * p.y;
        o4[i] = r;
    }
    for (long long i = (n4 << 2) + gid; i < n; i += gs)
        out[i] = __builtin_amdgcn_fmed3f(rintf(x[i] * p.x), p.z, p.w) * p.y;
}

// ---------- host ----------
extern "C" void kernel_launch(void* const* d_in, const int* in_sizes, int n_in,
                              void* d_out, int out_size, void* d_ws, size_t ws_size,
                              hipStream_t stream) {
    const float* x = (const float*)d_in[0];
    long long n = (long long)in_sizes[0];
    float* out = (float*)d_out;

    char* ws = (char*)d_ws;
    int*    keys   = (int*)(ws + 0);      //  8 B
    float4* best   = (float4*)(ws + 16);  // 16 B
    float*  mses   = (float*)(ws + 32);   // 320 B
    float4* params = (float4*)(ws + 384); // 1280 B

    k_init<<<1, 128, 0, stream>>>(keys, mses);
    k_minmax<<<STR_BLOCKS, STR_THREADS, 0, stream>>>(x, n, keys);
    k_params<<<1, 128, 0, stream>>>(keys, params);
    k_mse<<<MSE_BLOCKS, MSE_THREADS, 0, stream>>>(x, n, params, mses);
    k_best<<<1, 32, 0, stream>>>(mses, params, best);
    k_apply<<<STR_BLOCKS, STR_THREADS, 0, stream>>>(x, out, n, best);
}